// TrianguLangSheafLoss_26173530701951
// MI455X (gfx1250) — compile-verified
//
#include <hip/hip_runtime.h>
#include <math.h>

typedef float v2f __attribute__((ext_vector_type(2)));
typedef float v8f __attribute__((ext_vector_type(8)));

#define HWSZ 65536
#define DFEAT 64
#define TEMP_INV (1.0f / 0.07f)
#define THR_CONTR_F 0.1f
#define THR_CYCLE_F 0.05f

// workspace layout (bytes)
#define OFF_KEYS_CONTR 0u        // 2048 u64  (keys_contr + keys_ij + keys_jk contiguous: 3072 u64)
#define OFF_KEYS_IJ    16384u    // 512 u64
#define OFF_KEYS_JK    20480u    // 512 u64
#define OFF_EXPSUM     24576u    // 2048 f32
#define OFF_POSSIM     32768u    // 2048 f32
#define OFF_SE         40960u    // 512 f32
#define OFF_VALIDC     43008u    // 512 f32
#define OFF_INVNORM    45056u    // 786432 f32 (12 maps x 65536 rows)

// ---------------------------------------------------------------- init
__global__ void init_ws(unsigned long long* __restrict__ keys,
                        float* __restrict__ expsum) {
  int i = blockIdx.x * blockDim.x + threadIdx.x;
  if (i < 3072) keys[i] = 0xFFFFFFFFFFFFFFFFull;
  else if (i < 3072 + 2048) expsum[i - 3072] = 0.0f;
}

// ---------------------------------------------------------------- row norms
__global__ void norms_kernel(const float* __restrict__ f,
                             float* __restrict__ invnorm) {
  int row = blockIdx.x * blockDim.x + threadIdx.x;  // 0 .. 786431
  if (row >= 12 * HWSZ) return;
  const float4* p = (const float4*)(f + (size_t)row * DFEAT);
  float s = 0.0f;
#pragma unroll
  for (int i = 0; i < 16; ++i) {
    float4 v = p[i];
    s += v.x * v.x + v.y * v.y + v.z * v.z + v.w * v.w;
  }
  invnorm[row] = 1.0f / fmaxf(sqrtf(s), 1e-12f);
}

// ---------------------------------------------------------------- NN argmin
// mode 0: contrastive  — job z in [0,8): b=z>>1, target view (z&1)+1, A=256,
//         anchors = p[b,0][aidx_contr]
// mode 1: cycle i->j   — job z in [0,4): b=z, src view 0, tgt view 1, A=128,
//         anchors = p[b,0][aidx_cycle]
// mode 2: cycle j->k   — b=z, src view 1, tgt view 2, A=128,
//         anchors = p[b,1][low32(prev_keys)]
#define CHUNK 1024
__global__ void argmin_kernel(const float* __restrict__ pm,
                              const int* __restrict__ aidx_contr,
                              const int* __restrict__ aidx_cycle,
                              const unsigned long long* __restrict__ prev_keys,
                              unsigned long long* __restrict__ keys,
                              int mode) {
  // AoS point tile: (x,y,z) interleaved exactly as in memory so the async
  // DMA can copy it verbatim. 1024 points * 12B = 12KB, 16B aligned.
  __shared__ __align__(16) float sp[CHUNK * 3];
  int t = threadIdx.x;
  int job = blockIdx.y;
  int b, src_view, tgt_view, A;
  int arow = 0;
  if (mode == 0) {
    b = job >> 1; src_view = 0; tgt_view = (job & 1) + 1; A = 256;
    if (t < A) arow = aidx_contr[job * 256 + t];
  } else if (mode == 1) {
    b = job; src_view = 0; tgt_view = 1; A = 128;
    if (t < A) arow = aidx_cycle[b * 128 + t];
  } else {
    b = job; src_view = 1; tgt_view = 2; A = 128;
    if (t < A) arow = (int)(unsigned)(prev_keys[b * 128 + t] & 0xffffffffull);
  }
  const float* psrc = pm + (size_t)(b * 3 + src_view) * HWSZ * 3;
  const float* ptgt = pm + (size_t)(b * 3 + tgt_view) * HWSZ * 3;
  float ax = 0.f, ay = 0.f, az = 0.f;
  if (t < A) {
    ax = psrc[(size_t)arow * 3 + 0];
    ay = psrc[(size_t)arow * 3 + 1];
    az = psrc[(size_t)arow * 3 + 2];
  }
  int base = blockIdx.x * CHUNK;
  // Async DMA the 12KB point tile into LDS: 256 threads x 3 x b128.
  {
    unsigned ldsbase = (unsigned)(unsigned long long)(&sp[0]);
    const char* gsrc = (const char*)ptgt + (size_t)base * 12;
#pragma unroll
    for (int k = 0; k < 3; ++k) {
      unsigned la = ldsbase + (unsigned)(t * 16 + k * 4096);
      const char* ga = gsrc + t * 16 + k * 4096;
      asm volatile("global_load_async_to_lds_b128 %0, %1, off"
                   :: "v"(la), "v"(ga) : "memory");
    }
    asm volatile("s_wait_asynccnt 0" ::: "memory");
  }
  __syncthreads();
  if (t < A) {
    float best = 3.4e38f;
    int bidx = 0;
    for (int i = 0; i < CHUNK; ++i) {
      float dx = ax - sp[3 * i + 0];
      float dy = ay - sp[3 * i + 1];
      float dz = az - sp[3 * i + 2];
      float d = dx * dx + dy * dy + dz * dz;
      if (d < best) { best = d; bidx = base + i; }
    }
    unsigned long long key =
        (((unsigned long long)__float_as_uint(best)) << 32) | (unsigned)bidx;
    atomicMin(&keys[job * A + t], key);
  }
}

// ---------------------------------------------------------------- positive sim
__global__ void possim_kernel(const float* __restrict__ f,
                              const int* __restrict__ aidx_contr,
                              const unsigned long long* __restrict__ keys,
                              float* __restrict__ possim) {
  int gtid = blockIdx.x * blockDim.x + threadIdx.x;
  int wave = gtid >> 5;        // 0..2047
  int lane = threadIdx.x & 31;
  int job = wave >> 8, a = wave & 255;
  int b = job >> 1, v = (job & 1) + 1;
  const float* f0 = f + (size_t)(b * 3 + 0) * HWSZ * DFEAT;
  const float* fv = f + (size_t)(b * 3 + v) * HWSZ * DFEAT;
  int arow = aidx_contr[job * 256 + a];
  int prow = (int)(unsigned)(keys[job * 256 + a] & 0xffffffffull);
  float daa = 0.f, dpp = 0.f, dap = 0.f;
#pragma unroll
  for (int d = 0; d < 2; ++d) {
    int c = lane + d * 32;
    float x = f0[(size_t)arow * DFEAT + c];
    float y = fv[(size_t)prow * DFEAT + c];
    daa += x * x; dpp += y * y; dap += x * y;
  }
#pragma unroll
  for (int off = 16; off >= 1; off >>= 1) {
    daa += __shfl_xor(daa, off, 32);
    dpp += __shfl_xor(dpp, off, 32);
    dap += __shfl_xor(dap, off, 32);
  }
  if (lane == 0) {
    float na = fmaxf(sqrtf(daa), 1e-12f);
    float np = fmaxf(sqrtf(dpp), 1e-12f);
    possim[job * 256 + a] = (dap / (na * np)) * TEMP_INV;
  }
}

// ---------------------------------------------------------------- WMMA GEMM + exp-sum
// grid (32 n-chunks of 2048 pts, 4 m-groups of 64 anchors, 8 jobs), block = 1 wave.
// Each wave holds A fragments for 4 anchor tiles (64 anchors, K=64) in registers
// and reuses every B fragment across 4 independent WMMA accumulation chains,
// cutting f_j (L2) traffic 4x vs one-tile-per-wave.
// __launch_bounds__(32, 1): single-wave workgroup, 1 wave/EU minimum -> full
// VGPR budget so the ~212-VGPR working set (A frags + 4 C tiles + exp accs)
// stays in registers (round 2 spilled to scratch without this).
__global__ void __launch_bounds__(32, 1)
gemm_expsum_kernel(const float* __restrict__ f,
                   const int* __restrict__ aidx_contr,
                   const float* __restrict__ invnorm,
                   float* __restrict__ expsum) {
  int l = threadIdx.x;            // 0..31
  int nchunk = blockIdx.x;        // 0..31
  int mg = blockIdx.y;            // 0..3  (4 tiles of 16 anchors each)
  int job = blockIdx.z;           // 0..7
  int b = job >> 1, v = (job & 1) + 1;
  const float* f0 = f + (size_t)(b * 3 + 0) * HWSZ * DFEAT;
  const float* fv = f + (size_t)(b * 3 + v) * HWSZ * DFEAT;
  const float* invn0 = invnorm + (size_t)(b * 3 + 0) * HWSZ;
  const float* invnv = invnorm + (size_t)(b * 3 + v) * HWSZ;
  int h = l >> 4, ln = l & 15;

  // A fragments: 4 tiles x 16 normalized anchor rows, K=64 in 16 steps of 4
  v2f Afrag[4][16];
#pragma unroll
  for (int mt = 0; mt < 4; ++mt) {
    int arow = aidx_contr[job * 256 + (mg * 4 + mt) * 16 + ln];
    float ia = invn0[arow];
    const float* arp = f0 + (size_t)arow * DFEAT;
#pragma unroll
    for (int kk = 0; kk < 16; ++kk) {
      float2 tv = *(const float2*)(arp + kk * 4 + 2 * h);
      v2f av = {tv.x * ia, tv.y * ia};
      Afrag[mt][kk] = av;
    }
  }

  float acc[4][8];
#pragma unroll
  for (int mt = 0; mt < 4; ++mt)
#pragma unroll
    for (int r = 0; r < 8; ++r) acc[mt][r] = 0.0f;

  int nbase0 = nchunk * 2048;
  for (int nt = 0; nt < 128; ++nt) {
    int nbase = nbase0 + nt * 16;
    int prow = nbase + ln;
    const float* fr = fv + (size_t)prow * DFEAT;
    v8f C0 = {0, 0, 0, 0, 0, 0, 0, 0};
    v8f C1 = {0, 0, 0, 0, 0, 0, 0, 0};
    v8f C2 = {0, 0, 0, 0, 0, 0, 0, 0};
    v8f C3 = {0, 0, 0, 0, 0, 0, 0, 0};
#pragma unroll
    for (int kk = 0; kk < 16; ++kk) {
      float2 tv = *(const float2*)(fr + kk * 4 + 2 * h);
      v2f Bv = {tv.x, tv.y};
      C0 = __builtin_amdgcn_wmma_f32_16x16x4_f32(
          false, Afrag[0][kk], false, Bv, (short)0, C0, false, false);
      C1 = __builtin_amdgcn_wmma_f32_16x16x4_f32(
          false, Afrag[1][kk], false, Bv, (short)0, C1, false, false);
      C2 = __builtin_amdgcn_wmma_f32_16x16x4_f32(
          false, Afrag[2][kk], false, Bv, (short)0, C2, false, false);
      C3 = __builtin_amdgcn_wmma_f32_16x16x4_f32(
          false, Afrag[3][kk], false, Bv, (short)0, C3, false, false);
    }
    float inj = invnv[prow] * TEMP_INV;   // column scale (depends on n = ln)
#pragma unroll
    for (int r = 0; r < 8; ++r) {
      acc[0][r] += __expf(C0[r] * inj);
      acc[1][r] += __expf(C1[r] * inj);
      acc[2][r] += __expf(C2[r] * inj);
      acc[3][r] += __expf(C3[r] * inj);
    }
  }

  // reduce across the 16 lanes that share a row, then atomic into expsum
#pragma unroll
  for (int mt = 0; mt < 4; ++mt) {
#pragma unroll
    for (int r = 0; r < 8; ++r) {
      float vsum = acc[mt][r];
#pragma unroll
      for (int off = 8; off >= 1; off >>= 1) vsum += __shfl_xor(vsum, off, 32);
      if (ln == 0) {
        int m = (mg * 4 + mt) * 16 + r + 8 * h;
        atomicAdd(&expsum[job * 256 + m], vsum);
      }
    }
  }
}

// ---------------------------------------------------------------- cycle tail
__global__ void cycle_kernel(const float* __restrict__ f,
                             const float* __restrict__ pm,
                             const int* __restrict__ aidx_cycle,
                             const unsigned long long* __restrict__ keys_ij,
                             const unsigned long long* __restrict__ keys_jk,
                             float* __restrict__ se_out,
                             float* __restrict__ valid_out) {
  int t = blockIdx.x * blockDim.x + threadIdx.x;  // 0..511
  if (t >= 512) return;
  int b = t >> 7, a = t & 127;
  const float* p0 = pm + (size_t)(b * 3 + 0) * HWSZ * 3;
  const float* p2 = pm + (size_t)(b * 3 + 2) * HWSZ * 3;
  const float* f0 = f + (size_t)(b * 3 + 0) * HWSZ * DFEAT;
  unsigned long long kij = keys_ij[b * 128 + a];
  unsigned long long kjk = keys_jk[b * 128 + a];
  float md_ij = sqrtf(__uint_as_float((unsigned)(kij >> 32)));
  float md_jk = sqrtf(__uint_as_float((unsigned)(kjk >> 32)));
  int idx_k = (int)(unsigned)(kjk & 0xffffffffull);
  float kx = p2[(size_t)idx_k * 3 + 0];
  float ky = p2[(size_t)idx_k * 3 + 1];
  float kz = p2[(size_t)idx_k * 3 + 2];
  float best = 3.4e38f;
  int bi = 0;
  for (int a2 = 0; a2 < 128; ++a2) {
    int r = aidx_cycle[b * 128 + a2];
    float dx = kx - p0[(size_t)r * 3 + 0];
    float dy = ky - p0[(size_t)r * 3 + 1];
    float dz = kz - p0[(size_t)r * 3 + 2];
    float d = dx * dx + dy * dy + dz * dz;
    if (d < best) { best = d; bi = a2; }
  }
  float md_ki = sqrtf(best);
  float valid =
      (md_ij < THR_CYCLE_F && md_jk < THR_CYCLE_F && md_ki < THR_CYCLE_F) ? 1.0f : 0.0f;
  int arow = aidx_cycle[b * 128 + a];
  const float* fr = f0 + (size_t)bi * DFEAT;   // faithful: idx_ret indexes full map
  const float* fo = f0 + (size_t)arow * DFEAT;
  float srr = 0.f, soo = 0.f, sro = 0.f;
  for (int d = 0; d < DFEAT; ++d) {
    float x = fr[d], y = fo[d];
    srr += x * x; soo += y * y; sro += x * y;
  }
  float nr = fmaxf(sqrtf(srr), 1e-12f);
  float no = fmaxf(sqrtf(soo), 1e-12f);
  float se = srr / (nr * nr) + soo / (no * no) - 2.0f * sro / (nr * no);
  se_out[t] = se;
  valid_out[t] = valid;
}

// ---------------------------------------------------------------- final reduction
__global__ void final_kernel(const unsigned long long* __restrict__ keys_contr,
                             const float* __restrict__ possim,
                             const float* __restrict__ expsum,
                             const float* __restrict__ se,
                             const float* __restrict__ validc,
                             float* __restrict__ out) {
  __shared__ float s_l[256];
  __shared__ float s_c[256];
  int t = threadIdx.x;
  float contr = 0.0f;
  for (int v = 0; v < 2; ++v) {
    float lsum = 0.0f, wsum = 0.0f;
    for (int b = 0; b < 4; ++b) {
      int job = b * 2 + v;
      unsigned long long k = keys_contr[job * 256 + t];
      float md = sqrtf(__uint_as_float((unsigned)(k >> 32)));
      float vld = (md < THR_CONTR_F) ? 1.0f : 0.0f;
      float lv = -possim[job * 256 + t] + logf(expsum[job * 256 + t]);
      s_l[t] = lv * vld;
      s_c[t] = vld;
      __syncthreads();
      for (int off = 128; off >= 1; off >>= 1) {
        if (t < off) { s_l[t] += s_l[t + off]; s_c[t] += s_c[t + off]; }
        __syncthreads();
      }
      float cnt = s_c[0];
      float loss = s_l[0] / fmaxf(cnt, 1.0f);
      float w = (cnt >= 5.0f) ? 1.0f : 0.0f;
      lsum += loss * w;
      wsum += w;
      __syncthreads();
    }
    contr += (wsum > 0.0f) ? (lsum / fmaxf(wsum, 1.0f)) : 0.0f;
  }
  contr *= 0.5f;  // / (N-1)

  float clsum = 0.0f, cwsum = 0.0f;
  for (int b = 0; b < 4; ++b) {
    if (t < 128) {
      s_l[t] = se[b * 128 + t] * validc[b * 128 + t];
      s_c[t] = validc[b * 128 + t];
    } else { s_l[t] = 0.0f; s_c[t] = 0.0f; }
    __syncthreads();
    for (int off = 128; off >= 1; off >>= 1) {
      if (t < off) { s_l[t] += s_l[t + off]; s_c[t] += s_c[t + off]; }
      __syncthreads();
    }
    float cnt = s_c[0];
    float loss = s_l[0] / (fmaxf(cnt, 1.0f) * (float)DFEAT);
    float w = (cnt >= 5.0f) ? 1.0f : 0.0f;
    clsum += loss * w;
    cwsum += w;
    __syncthreads();
  }
  float cyc = (cwsum > 0.0f) ? (clsum / fmaxf(cwsum, 1.0f)) : 0.0f;  // n_trip = 1
  if (t == 0) out[0] = contr + cyc;
}

// ---------------------------------------------------------------- launch
extern "C" void kernel_launch(void* const* d_in, const int* in_sizes, int n_in,
                              void* d_out, int out_size, void* d_ws, size_t ws_size,
                              hipStream_t stream) {
  (void)in_sizes; (void)n_in; (void)out_size; (void)ws_size;
  const float* f = (const float*)d_in[0];
  const float* pm = (const float*)d_in[1];
  const int* aidx_contr = (const int*)d_in[2];
  const int* aidx_cycle = (const int*)d_in[3];
  float* out = (float*)d_out;
  char* ws = (char*)d_ws;
  unsigned long long* keys_contr = (unsigned long long*)(ws + OFF_KEYS_CONTR);
  unsigned long long* keys_ij = (unsigned long long*)(ws + OFF_KEYS_IJ);
  unsigned long long* keys_jk = (unsigned long long*)(ws + OFF_KEYS_JK);
  float* expsum = (float*)(ws + OFF_EXPSUM);
  float* possim = (float*)(ws + OFF_POSSIM);
  float* se = (float*)(ws + OFF_SE);
  float* validc = (float*)(ws + OFF_VALIDC);
  float* invnorm = (float*)(ws + OFF_INVNORM);

  init_ws<<<20, 256, 0, stream>>>(keys_contr, expsum);
  norms_kernel<<<3072, 256, 0, stream>>>(f, invnorm);
  argmin_kernel<<<dim3(64, 8), 256, 0, stream>>>(pm, aidx_contr, aidx_cycle,
                                                 nullptr, keys_contr, 0);
  argmin_kernel<<<dim3(64, 4), 256, 0, stream>>>(pm, aidx_contr, aidx_cycle,
                                                 nullptr, keys_ij, 1);
  argmin_kernel<<<dim3(64, 4), 256, 0, stream>>>(pm, aidx_contr, aidx_cycle,
                                                 keys_ij, keys_jk, 2);
  possim_kernel<<<256, 256, 0, stream>>>(f, aidx_contr, keys_contr, possim);
  gemm_expsum_kernel<<<dim3(32, 4, 8), 32, 0, stream>>>(f, aidx_contr, invnorm,
                                                        expsum);
  cycle_kernel<<<2, 256, 0, stream>>>(f, pm, aidx_cycle, keys_ij, keys_jk, se,
                                      validc);
  final_kernel<<<1, 256, 0, stream>>>(keys_contr, possim, expsum, se, validc,
                                      out);
}